// SparseMoEFFN_44341242364491
// MI455X (gfx1250) — compile-verified
//
#include <hip/hip_runtime.h>

// Sparse MoE FFN for MI455X (gfx1250, wave32).
// T=8192 tokens, H=1024, EH=4096, E=8 experts, top-2 routing.
// Weight traffic (268 MB fp32) is the roofline bound (~13us @ 23.3 TB/s), so:
//  - fp32 WMMA (V_WMMA_F32_16X16X4_F32) keeps reference precision at no cost
//  - fused fc1->silu->fc2 keeps the h intermediate entirely in LDS (128 KB/WG,
//    legal on CDNA5's 320 KB WGP LDS) so h never touches HBM.
// Round-1 fixes: silu via v_rcp_f32 instead of full IEEE div expansion;
// deeper K unroll (8) to widen the load/WMMA clauses.

#define H      1024
#define EH     4096
#define NE     8
#define T_TOK  8192
#define CAP    8192    // per-expert token capacity (a token appears at most once per expert)
#define MT     16      // tokens per FFN block (one WMMA M tile)
#define CHUNK  1024    // EH processed per LDS-resident h chunk

typedef __attribute__((ext_vector_type(2))) float v2f;
typedef __attribute__((ext_vector_type(8))) float v8f;

__device__ __forceinline__ v8f wmma_f32(v2f a, v2f b, v8f c) {
  // 8 args: (neg_a, A, neg_b, B, c_mod, C, reuse_a, reuse_b)
  return __builtin_amdgcn_wmma_f32_16x16x4_f32(false, a, false, b, (short)0, c,
                                               false, false);
}

__device__ __forceinline__ float fast_rcp(float x) {
  return __builtin_amdgcn_rcpf(x);  // single v_rcp_f32, ~1 ulp
}

// ---------------------------------------------------------------------------
// Kernel 1: zero output + per-expert assignment counters
// ---------------------------------------------------------------------------
__global__ void moe_zero(float* __restrict__ out, int n, int* __restrict__ counts) {
  int i = blockIdx.x * blockDim.x + threadIdx.x;
  if (i < n) out[i] = 0.f;
  if (i < NE) counts[i] = 0;
}

// ---------------------------------------------------------------------------
// Kernel 2: router. One wave32 per token: lane-parallel dot products over H,
// shuffle-reduce, softmax over 8 logits, top-2 + renorm, append to expert list.
// ---------------------------------------------------------------------------
__global__ __launch_bounds__(256)
void moe_router(const float* __restrict__ x, const float* __restrict__ rw,
                int* __restrict__ counts, int* __restrict__ tok,
                float* __restrict__ wgt) {
  int wave = threadIdx.x >> 5;
  int lane = threadIdx.x & 31;
  int t = blockIdx.x * 8 + wave;
  if (t >= T_TOK) return;

  float acc[NE];
#pragma unroll
  for (int e = 0; e < NE; ++e) acc[e] = 0.f;

  for (int k = lane; k < H; k += 32) {
    float xv = x[t * H + k];
#pragma unroll
    for (int e = 0; e < NE; ++e) acc[e] += xv * rw[e * H + k];
  }
#pragma unroll
  for (int e = 0; e < NE; ++e) {
#pragma unroll
    for (int off = 16; off >= 1; off >>= 1)
      acc[e] += __shfl_xor(acc[e], off, 32);
  }

  if (lane == 0) {
    float mx = acc[0];
#pragma unroll
    for (int e = 1; e < NE; ++e) mx = fmaxf(mx, acc[e]);
    float p[NE];
    float s = 0.f;
#pragma unroll
    for (int e = 0; e < NE; ++e) { p[e] = __expf(acc[e] - mx); s += p[e]; }
    float inv = fast_rcp(s);
#pragma unroll
    for (int e = 0; e < NE; ++e) p[e] *= inv;

    int i0 = 0; float p0 = p[0];
#pragma unroll
    for (int e = 1; e < NE; ++e) if (p[e] > p0) { p0 = p[e]; i0 = e; }
    int i1 = (i0 == 0) ? 1 : 0; float p1 = p[i1];
#pragma unroll
    for (int e = 0; e < NE; ++e)
      if (e != i0 && p[e] > p1) { p1 = p[e]; i1 = e; }

    float rden = fast_rcp(fmaxf(p0 + p1, 1e-9f));
    float w0 = p0 * rden, w1 = p1 * rden;

    int s0 = atomicAdd(&counts[i0], 1);
    tok[i0 * CAP + s0] = t; wgt[i0 * CAP + s0] = w0;
    int s1 = atomicAdd(&counts[i1], 1);
    tok[i1 * CAP + s1] = t; wgt[i1 * CAP + s1] = w1;
  }
}

// ---------------------------------------------------------------------------
// Kernel 3: fused expert FFN. Block = 256 threads (8 waves), 16 tokens.
//   A[16,1024] gathered to LDS once.
//   For each EH chunk of 1024: h = silu(A @ W1^T) computed by WMMA into LDS,
//   then y += h @ W2chunk^T with y in registers (16x1024 per block).
//   Combine: out[t] += w * y via float atomics (exactly 2 commutative adds
//   per element -> bit-deterministic).
// WMMA fragment maps (V_WMMA_F32_16X16X4_F32, wave32):
//   A 16x4 : lane l -> M = l&15, K = k0 + 2*(l>>4) + {vgpr0,vgpr1}
//   B 4x16 : lane l -> N = l&15, K mirrored like A
//   C 16x16: lane l, vgpr v -> N = l&15, M = v + 8*(l>>4)
// ---------------------------------------------------------------------------
__global__ __launch_bounds__(256)
void moe_ffn(const float* __restrict__ x, const float* __restrict__ w1,
             const float* __restrict__ w2, const int* __restrict__ counts,
             const int* __restrict__ tok, const float* __restrict__ wgt,
             float* __restrict__ out) {
  extern __shared__ float smem[];
  float* a_lds = smem;            // MT * H     = 64 KB
  float* h_lds = smem + MT * H;   // MT * CHUNK = 64 KB
  __shared__ int   tok16[MT];
  __shared__ float w16[MT];

  const int e = blockIdx.y;
  const int base = blockIdx.x * MT;
  const int cnt = counts[e];
  if (base >= cnt) return;

  const int tid = threadIdx.x;
  const int lane = tid & 31;
  const int wave = tid >> 5;
  const int l16 = lane & 15;           // M for A-frag, N for B/C-frag
  const int khalf = (lane >> 4) << 1;  // 0 or 2

  if (tid < MT) {
    int idx = base + tid;
    tok16[tid] = (idx < cnt) ? tok[e * CAP + idx] : 0;
    w16[tid]   = (idx < cnt) ? wgt[e * CAP + idx] : 0.f;
  }
  __syncthreads();

  // Gather activation tile; pad rows with zeros (silu(0)=0, store is masked).
  for (int i = tid; i < MT * H; i += 256) {
    int m = i >> 10;
    int k = i & (H - 1);
    a_lds[i] = (base + m < cnt) ? x[tok16[m] * H + k] : 0.f;
  }
  __syncthreads();

  const float* w1e = w1 + (size_t)e * EH * H;
  const float* w2e = w2 + (size_t)e * H * EH;

  v8f yacc[8] = {};  // this wave owns y columns [wave*128, wave*128+128)

  for (int c = 0; c < EH / CHUNK; ++c) {
    // ---- Phase A: h_lds[0:16, 0:CHUNK] = silu(A @ W1_chunk^T) ----
#pragma unroll 1
    for (int i = 0; i < (CHUNK / 128); ++i) {  // 8 N-tiles per wave
      const int nloc = wave * (CHUNK / 8) + i * 16;
      const int nglb = c * CHUNK + nloc;
      const float* brow = w1e + (size_t)(nglb + l16) * H;  // W1 row for this lane's N
      v8f acc = {};
#pragma unroll 8
      for (int k0 = 0; k0 < H; k0 += 4) {
        v2f a, b;
        a.x = a_lds[l16 * H + k0 + khalf];
        a.y = a_lds[l16 * H + k0 + khalf + 1];
        b.x = brow[k0 + khalf];
        b.y = brow[k0 + khalf + 1];
        acc = wmma_f32(a, b, acc);
      }
#pragma unroll
      for (int v = 0; v < 8; ++v) {
        int m = v + ((lane >> 4) << 3);
        float hv = acc[v];
        hv = hv * fast_rcp(1.f + __expf(-hv));  // silu via v_rcp_f32
        h_lds[m * CHUNK + nloc + l16] = hv;
      }
    }
    __syncthreads();

    // ---- Phase B: y += h_chunk @ W2_chunk^T ----
#pragma unroll 1
    for (int j = 0; j < 8; ++j) {
      const int n2 = wave * 128 + j * 16 + l16;  // output column (0..1023)
      const float* brow2 = w2e + (size_t)n2 * EH + c * CHUNK;
      v8f acc = yacc[j];
#pragma unroll 8
      for (int k0 = 0; k0 < CHUNK; k0 += 4) {
        v2f a, b;
        a.x = h_lds[l16 * CHUNK + k0 + khalf];
        a.y = h_lds[l16 * CHUNK + k0 + khalf + 1];
        b.x = brow2[k0 + khalf];
        b.y = brow2[k0 + khalf + 1];
        acc = wmma_f32(a, b, acc);
      }
      yacc[j] = acc;
    }
    __syncthreads();  // before next chunk overwrites h_lds
  }

  // ---- Combine: out[t] += w * y (two commutative adds per element) ----
#pragma unroll
  for (int j = 0; j < 8; ++j) {
    int n = wave * 128 + j * 16 + l16;
#pragma unroll
    for (int v = 0; v < 8; ++v) {
      int m = v + ((lane >> 4) << 3);
      if (base + m < cnt)
        atomicAdd(&out[(size_t)tok16[m] * H + n], w16[m] * yacc[j][v]);
    }
  }
}

// ---------------------------------------------------------------------------
extern "C" void kernel_launch(void* const* d_in, const int* in_sizes, int n_in,
                              void* d_out, int out_size, void* d_ws, size_t ws_size,
                              hipStream_t stream) {
  const float* x  = (const float*)d_in[0];  // [4,2048,1024]
  const float* rw = (const float*)d_in[1];  // [8,1024]
  const float* w1 = (const float*)d_in[2];  // [8,4096,1024]
  const float* w2 = (const float*)d_in[3];  // [8,1024,4096]
  float* out = (float*)d_out;               // [4,2048,1024]

  char* ws = (char*)d_ws;
  int*   counts = (int*)ws;                                  // 8 ints
  int*   tok    = (int*)(ws + 128);                          // 8*8192 ints
  float* wgt    = (float*)(ws + 128 + NE * CAP * sizeof(int));

  moe_zero<<<(out_size + 255) / 256, 256, 0, stream>>>(out, out_size, counts);
  moe_router<<<T_TOK / 8, 256, 0, stream>>>(x, rw, counts, tok, wgt);

  dim3 grid(T_TOK / MT, NE);  // most blocks early-exit on counts[e]
  size_t shmem = (size_t)(MT * H + MT * CHUNK) * sizeof(float);  // 128 KB
  moe_ffn<<<grid, 256, shmem, stream>>>(x, w1, w2, counts, tok, wgt, out);
}